// PPIEncoder3_36447092474375
// MI455X (gfx1250) — compile-verified
//
#include <hip/hip_runtime.h>
#include <hip/hip_bf16.h>

typedef __attribute__((ext_vector_type(2))) float v2f;
typedef __attribute__((ext_vector_type(8))) float v8f;

// ---------------------------------------------------------------------------
// Utility: zero a float buffer
// ---------------------------------------------------------------------------
__global__ void zero_kernel(float* __restrict__ p, size_t n) {
    size_t i = (size_t)blockIdx.x * blockDim.x + threadIdx.x;
    if (i < n) p[i] = 0.0f;
}

// ---------------------------------------------------------------------------
// Degree count: cnt[dst[e]] += 1
// ---------------------------------------------------------------------------
__global__ void deg_kernel(const int* __restrict__ dst, float* __restrict__ cnt, int nE) {
    int e = blockIdx.x * blockDim.x + threadIdx.x;
    if (e < nE) atomicAdd(&cnt[dst[e]], 1.0f);
}

// cnt[i] <- 1 / max(cnt[i], 1)
__global__ void inv_kernel(float* __restrict__ cnt, int n) {
    int i = blockIdx.x * blockDim.x + threadIdx.x;
    if (i < n) cnt[i] = 1.0f / fmaxf(cnt[i], 1.0f);
}

// ---------------------------------------------------------------------------
// Scatter-add: agg[dst[e]] += h[src[e]]   (64 features per edge)
// 16 threads per edge; each thread handles 4 consecutive features (float4
// gather is coalesced across the 16 lanes sharing an edge).
// ---------------------------------------------------------------------------
__global__ void scatter_add_kernel(const float* __restrict__ h,
                                   const int* __restrict__ src,
                                   const int* __restrict__ dst,
                                   float* __restrict__ agg, int nE) {
    int t = blockIdx.x * blockDim.x + threadIdx.x;
    int e = t >> 4;
    int c = (t & 15) << 2;
    if (e >= nE) return;
    int s = src[e];
    int d = dst[e];
    const float4 v = *(const float4*)(h + (size_t)s * 64 + c);
    float* p = agg + (size_t)d * 64 + c;
    atomicAdd(p + 0, v.x);
    atomicAdd(p + 1, v.y);
    atomicAdd(p + 2, v.z);
    atomicAdd(p + 3, v.w);
}

// agg[i][f] *= inv[i]   -> mean
__global__ void scale_kernel(float* __restrict__ agg, const float* __restrict__ inv, size_t n64) {
    size_t i = (size_t)blockIdx.x * blockDim.x + threadIdx.x;
    if (i < n64) agg[i] *= inv[i >> 6];
}

// ---------------------------------------------------------------------------
// Fused SAGE transform using dense f32 WMMA (v_wmma_f32_16x16x4_f32):
//   out = act( mean @ Wl^T + b + h @ Wr^T )
// One wave computes a 16x16 output tile; K=64 per operand -> 32 WMMA ops.
// A-fragment (16x4 f32): lanes 0-15 hold K=kk,kk+1; lanes 16-31 hold K=kk+2,kk+3.
// B-fragment (4x16 f32): lane n in 0-15 holds B[kk..kk+1][n]; lanes 16-31 hold
// B[kk+2..kk+3][n], where B[k][n] = W[n][k] (W is row-major [DOUT,64]).
// C/D layout: vgpr v, lane L: row = v + 8*(L>=16), col = L%16.
// ---------------------------------------------------------------------------
template <int DOUT, bool RELU>
__global__ __launch_bounds__(32 * (DOUT / 16))
void sage_gemm_kernel(const float* __restrict__ mean,  // [n,64] (pre-scaled)
                      const float* __restrict__ h,     // [n,64]
                      const float* __restrict__ Wl,    // [DOUT,64]
                      const float* __restrict__ bias,  // [DOUT]
                      const float* __restrict__ Wr,    // [DOUT,64]
                      float* __restrict__ out,         // [n,DOUT]
                      int n) {
    constexpr int K = 64;
    const int wave = threadIdx.x >> 5;
    const int lane = threadIdx.x & 31;
    const int lm   = lane & 15;
    const int hi   = lane >> 4;       // 0 or 1
    const int row0 = blockIdx.x * 16;
    const int col0 = wave * 16;

    int arow = row0 + lm;
    if (arow >= n) arow = n - 1;      // address clamp; EXEC stays all-ones
    const int bcol = col0 + lm;       // output feature index, always < DOUT
    const int koff = hi * 2;

    const float* Am = mean + (size_t)arow * K + koff;
    const float* Ah = h    + (size_t)arow * K + koff;
    const float* Bl = Wl   + (size_t)bcol * K + koff;
    const float* Br = Wr   + (size_t)bcol * K + koff;

    v8f c = {};
#pragma unroll
    for (int kk = 0; kk < K; kk += 4) {
        v2f a = *(const v2f*)(Am + kk);
        v2f b = *(const v2f*)(Bl + kk);
        c = __builtin_amdgcn_wmma_f32_16x16x4_f32(false, a, false, b,
                                                  (short)0, c, false, false);
    }
#pragma unroll
    for (int kk = 0; kk < K; kk += 4) {
        v2f a = *(const v2f*)(Ah + kk);
        v2f b = *(const v2f*)(Br + kk);
        c = __builtin_amdgcn_wmma_f32_16x16x4_f32(false, a, false, b,
                                                  (short)0, c, false, false);
    }

    const float bv = bias[bcol];
#pragma unroll
    for (int v = 0; v < 8; ++v) {
        int r = row0 + v + hi * 8;
        if (r < n) {
            float val = c[v] + bv;
            if (RELU) val = fmaxf(val, 0.0f);
            out[(size_t)r * DOUT + bcol] = val;
        }
    }
}

// ---------------------------------------------------------------------------
// Host-side orchestration
// ---------------------------------------------------------------------------
static inline int blks(size_t n, int t) { return (int)((n + t - 1) / t); }

extern "C" void kernel_launch(void* const* d_in, const int* in_sizes, int n_in,
                              void* d_out, int out_size, void* d_ws, size_t ws_size,
                              hipStream_t stream) {
    const float* x   = (const float*)d_in[0];
    const int*   ei  = (const int*)d_in[1];   // jnp.int64 w/o x64 -> int32
    const float* W1l = (const float*)d_in[2];
    const float* b1  = (const float*)d_in[3];
    const float* W1r = (const float*)d_in[4];
    const float* W2l = (const float*)d_in[5];
    const float* b2  = (const float*)d_in[6];
    const float* W2r = (const float*)d_in[7];
    const float* Wml = (const float*)d_in[8];
    const float* bm  = (const float*)d_in[9];
    const float* Wmr = (const float*)d_in[10];
    const float* Wsl = (const float*)d_in[11];
    const float* bs  = (const float*)d_in[12];
    const float* Wsr = (const float*)d_in[13];

    const int N = in_sizes[0] / 64;
    const int E = in_sizes[1] / 2;
    const int* src = ei;
    const int* dst = ei + E;

    const size_t NF   = (size_t)N * 64;
    const size_t npad = ((size_t)N + 15) & ~(size_t)15;
    float* ws   = (float*)d_ws;
    float* inv  = ws;              // [N]      degree counts -> 1/max(deg,1)
    float* bufA = ws + npad;       // [N,64]   h1
    float* bufB = bufA + NF;       // [N,64]   agg / mean scratch
    float* bufC = bufB + NF;       // [N,64]   h2

    float* mu = (float*)d_out;
    float* ls = mu + (size_t)N * 32;

    const int T = 256;
    const int n16 = (N + 15) / 16;

    // degrees -> reciprocal
    zero_kernel<<<blks(N, T), T, 0, stream>>>(inv, (size_t)N);
    deg_kernel<<<blks(E, T), T, 0, stream>>>(dst, inv, E);
    inv_kernel<<<blks(N, T), T, 0, stream>>>(inv, N);

    // ---- layer 1: h1 = relu(mean(x) @ W1l^T + b1 + x @ W1r^T) ----
    zero_kernel<<<blks(NF, T), T, 0, stream>>>(bufB, NF);
    scatter_add_kernel<<<blks((size_t)E * 16, T), T, 0, stream>>>(x, src, dst, bufB, E);
    scale_kernel<<<blks(NF, T), T, 0, stream>>>(bufB, inv, NF);
    sage_gemm_kernel<64, true><<<n16, 128, 0, stream>>>(bufB, x, W1l, b1, W1r, bufA, N);

    // ---- layer 2: h2 = relu(mean(h1) @ W2l^T + b2 + h1 @ W2r^T) ----
    zero_kernel<<<blks(NF, T), T, 0, stream>>>(bufB, NF);
    scatter_add_kernel<<<blks((size_t)E * 16, T), T, 0, stream>>>(bufA, src, dst, bufB, E);
    scale_kernel<<<blks(NF, T), T, 0, stream>>>(bufB, inv, NF);
    sage_gemm_kernel<64, true><<<n16, 128, 0, stream>>>(bufB, bufA, W2l, b2, W2r, bufC, N);

    // ---- heads: share mean(h2) ----
    zero_kernel<<<blks(NF, T), T, 0, stream>>>(bufB, NF);
    scatter_add_kernel<<<blks((size_t)E * 16, T), T, 0, stream>>>(bufC, src, dst, bufB, E);
    scale_kernel<<<blks(NF, T), T, 0, stream>>>(bufB, inv, NF);
    sage_gemm_kernel<32, false><<<n16, 64, 0, stream>>>(bufB, bufC, Wml, bm, Wmr, mu, N);
    sage_gemm_kernel<32, false><<<n16, 64, 0, stream>>>(bufB, bufC, Wsl, bs, Wsr, ls, N);
}